// MGRU_ori_65678639891094
// MI455X (gfx1250) — compile-verified
//
#include <hip/hip_runtime.h>
#include <math.h>
#include <stdint.h>

// ---------------- problem constants ----------------
#define F_DIM 256
#define H_DIM 512
#define BB    512
#define TT    128
#define TF    (TT * F_DIM)          // 32768
#define TF3   (3 * TF)              // 98304  (row stride of input[b,3,T,F])
#define BTF   (BB * TT * F_DIM)     // 16777216
#define H3    (3 * H_DIM)           // 1536

// ---------------- GEMM tiling ----------------
#define BM 64
#define BN 64
#define KC 32
#define KP 36          // padded LDS row stride (16B aligned, conflict-free frags)
#define NTHREADS 128   // 4 waves; each wave owns a 32x32 output subtile

typedef __attribute__((__ext_vector_type__(2))) float v2f;
typedef __attribute__((__ext_vector_type__(8))) float v8f;

struct SegDesc {
  const float* A;   // A rows: C-row index; element A[(arow0+r)*as + k]
  int          as;  // A row stride
  const float* W;   // W rows: C-col index; element W[(wrow0+r)*ws + k]
  int          ws;  // W row stride
  int          klen;
};

// Issue one 64xKC A tile + 64xKC W tile as async global->LDS B128 copies.
// 8 async instructions per wave per chunk (4 for A, 4 for W).
__device__ __forceinline__ void issue_chunk(const SegDesc& s, int arow0, int wrow0,
                                            int kb, float* Asb, float* Wsb) {
  int tid = threadIdx.x;
  unsigned lA = (unsigned)(uintptr_t)Asb;   // low 32 bits of flat shared addr = LDS offset
  unsigned lW = (unsigned)(uintptr_t)Wsb;
#pragma unroll
  for (int it = 0; it < 4; ++it) {
    int idx = tid + it * NTHREADS;        // 0..511 (64 rows x 8 float4)
    int r   = idx >> 3;
    int c4  = (idx & 7) << 2;
    const float* ga = s.A + (size_t)(arow0 + r) * s.as + kb + c4;
    const float* gw = s.W + (size_t)(wrow0 + r) * s.ws + kb + c4;
    unsigned la = lA + (unsigned)(r * KP + c4) * 4u;
    unsigned lw = lW + (unsigned)(r * KP + c4) * 4u;
    asm volatile("global_load_async_to_lds_b128 %0, %1, off"
                 :: "v"(la), "v"(ga) : "memory");
    asm volatile("global_load_async_to_lds_b128 %0, %1, off"
                 :: "v"(lw), "v"(gw) : "memory");
  }
}

// One KC-deep chunk of fp32 WMMA on the wave's 32x32 subtile.
// A frag (16x4 f32): lanes 0-15 hold K=k0,k0+1 ; lanes 16-31 hold K=k0+2,k0+3.
// B frag (4x16 f32): same K split; B[k][n] sourced as W[n][k].
__device__ __forceinline__ void wmma_chunk(const float* As, const float* Ws, v8f acc[2][2]) {
  int lane = threadIdx.x & 31;
  int wave = threadIdx.x >> 5;
  int wm = (wave >> 1) << 5;
  int wn = (wave & 1) << 5;
  int lr  = lane & 15;
  int khi = (lane >= 16) ? 2 : 0;
#pragma unroll
  for (int k0 = 0; k0 < KC; k0 += 4) {
    int ka = k0 + khi;
#pragma unroll
    for (int i = 0; i < 2; ++i) {
      const float* ap = As + (wm + 16 * i + lr) * KP + ka;
      v2f af; af.x = ap[0]; af.y = ap[1];
#pragma unroll
      for (int j = 0; j < 2; ++j) {
        const float* wp = Ws + (wn + 16 * j + lr) * KP + ka;
        v2f bf; bf.x = wp[0]; bf.y = wp[1];
        acc[i][j] = __builtin_amdgcn_wmma_f32_16x16x4_f32(
            false, af, false, bf, (short)0, acc[i][j], false, false);
      }
    }
  }
}

// Double-buffered async GEMM main loop.
// Prefetch chunk c+1 (8 async ops), then s_wait_asynccnt 8: async loads complete
// in order, so the 8 oldest (chunk c) have landed while the prefetch stays in flight.
template <bool ZDIAG>
__device__ __forceinline__ void gemm_run(const SegDesc* segs, int nseg, int arow0,
                                         int wrow0, v8f acc[2][2],
                                         float* As0, float* Ws0, float* As1, float* Ws1) {
  int n0  = segs[0].klen / KC;
  int nch = n0 + ((nseg > 1) ? segs[1].klen / KC : 0);

  issue_chunk(segs[0], arow0, wrow0, 0, As0, Ws0);
  for (int c = 0; c < nch; ++c) {
    bool more = (c + 1 < nch);
    if (more) {
      int cn = c + 1;
      int sn = (cn < n0) ? 0 : 1;
      int kbn = ((cn < n0) ? cn : (cn - n0)) * KC;
      issue_chunk(segs[sn], arow0, wrow0, kbn,
                  (cn & 1) ? As1 : As0, (cn & 1) ? Ws1 : Ws0);
      asm volatile("s_wait_asynccnt 0x8" ::: "memory");
    } else {
      asm volatile("s_wait_asynccnt 0x0" ::: "memory");
    }
    __syncthreads();
    float* Asb = (c & 1) ? As1 : As0;
    float* Wsb = (c & 1) ? Ws1 : Ws0;
    if (ZDIAG) {  // zero Wz diagonal in-place in the staged tile
      int kb = ((c < n0) ? c : (c - n0)) * KC;
      int tid = threadIdx.x;
      if (tid < KC) {
        int n  = kb + tid;      // square matrix: global k == global n on diagonal
        int rl = n - wrow0;
        if (rl >= 0 && rl < BN) Wsb[rl * KP + tid] = 0.f;
      }
      __syncthreads();
    }
    wmma_chunk(Asb, Wsb, acc);
    __syncthreads();
  }
}

// Iterate (row, region-local col, acc value) over this thread's 32 C elements.
// C/D layout: VGPR r -> lanes 0-15: (M=r, N=lane); lanes 16-31: (M=r+8, N=lane-16).
template <typename Fn>
__device__ __forceinline__ void tile_foreach(int bm, int nlb, v8f acc[2][2], Fn fn) {
  int lane = threadIdx.x & 31;
  int wave = threadIdx.x >> 5;
  int wm = (wave >> 1) << 5;
  int wn = (wave & 1) << 5;
  int lr = lane & 15;
  int ro = (lane >= 16) ? 8 : 0;
#pragma unroll
  for (int i = 0; i < 2; ++i)
#pragma unroll
    for (int j = 0; j < 2; ++j)
#pragma unroll
      for (int r = 0; r < 8; ++r)
        fn(bm + wm + 16 * i + r + ro, nlb + wn + 16 * j + lr, acc[i][j][r]);
}

__device__ __forceinline__ float sigmoidf_(float x) { return 1.f / (1.f + expf(-x)); }

#define GEMM_SHARED \
  __shared__ float As0[BM * KP]; __shared__ float Ws0[BN * KP]; \
  __shared__ float As1[BM * KP]; __shared__ float Ws1[BN * KP];

// ---------------- step kernels ----------------

// gamma_x = exp(-relu(d@Wdx^T+bdx))  [cols 0..255]
// hid_dec = hid * exp(-relu(d@Wdh^T+bdh))  [cols 256..767]
__global__ void k_gamma(const float* __restrict__ inp, int t,
                        const float* __restrict__ Wdx, const float* __restrict__ bdx,
                        const float* __restrict__ Wdh, const float* __restrict__ bdh,
                        const float* __restrict__ hid,
                        float* __restrict__ gamma_x, float* __restrict__ hid_dec) {
  GEMM_SHARED
  int bm = blockIdx.y * BM;
  int nb = blockIdx.x * BN;
  v8f acc[2][2] = {};
  const float* dbase = inp + TF + t * F_DIM;  // delta: row stride TF3
  if (nb < F_DIM) {
    SegDesc s[1] = {{dbase, TF3, Wdx, F_DIM, F_DIM}};
    gemm_run<false>(s, 1, bm, nb, acc, As0, Ws0, As1, Ws1);
    tile_foreach(bm, nb, acc, [&](int row, int col, float a) {
      gamma_x[row * F_DIM + col] = expf(-fmaxf(a + bdx[col], 0.f));
    });
  } else {
    int nl = nb - F_DIM;
    SegDesc s[1] = {{dbase, TF3, Wdh, F_DIM, F_DIM}};
    gemm_run<false>(s, 1, bm, nl, acc, As0, Ws0, As1, Ws1);
    tile_foreach(bm, nl, acc, [&](int row, int col, float a) {
      float g = expf(-fmaxf(a + bdh[col], 0.f));
      hid_dec[row * H_DIM + col] = hid[row * H_DIM + col] * g;
    });
  }
}

// cols 0..255: beta = [gamma_x, m]@Wbeta^T + bbeta
// cols 256..511: x_hat = hid_dec@Wout^T + bout ; x_c = m*v+(1-m)*x_hat ; loss1, mask sum
__global__ void k_beta_xhat(const float* __restrict__ inp, int t,
                            const float* __restrict__ gamma_x,
                            const float* __restrict__ hid_dec,
                            const float* __restrict__ Wbeta, const float* __restrict__ bbeta,
                            const float* __restrict__ Wout, const float* __restrict__ bout,
                            float* __restrict__ beta, float* __restrict__ x_hat,
                            float* __restrict__ x_c,
                            float* __restrict__ lossS, float* __restrict__ maskS) {
  GEMM_SHARED
  int bm = blockIdx.y * BM;
  int nb = blockIdx.x * BN;
  v8f acc[2][2] = {};
  const float* mbase = inp + 2 * TF + t * F_DIM;
  if (nb < F_DIM) {
    SegDesc s[2] = {{gamma_x, F_DIM, Wbeta, 2 * F_DIM, F_DIM},
                    {mbase, TF3, Wbeta + F_DIM, 2 * F_DIM, F_DIM}};
    gemm_run<false>(s, 2, bm, nb, acc, As0, Ws0, As1, Ws1);
    tile_foreach(bm, nb, acc, [&](int row, int col, float a) {
      beta[row * F_DIM + col] = a + bbeta[col];
    });
  } else {
    int nl = nb - F_DIM;
    SegDesc s[1] = {{hid_dec, H_DIM, Wout, H_DIM, H_DIM}};
    gemm_run<false>(s, 1, bm, nl, acc, As0, Ws0, As1, Ws1);
    float s1 = 0.f, sm = 0.f;
    tile_foreach(bm, nl, acc, [&](int row, int col, float a) {
      float xh = a + bout[col];
      float m  = inp[row * TF3 + 2 * TF + t * F_DIM + col];
      float vv = inp[row * TF3 + t * F_DIM + col];
      x_hat[row * F_DIM + col] = xh;
      x_c[row * F_DIM + col]   = m * vv + (1.f - m) * xh;
      s1 += fabsf(vv - xh) * m;
      sm += m;
    });
    atomicAdd(&lossS[t], s1);
    atomicAdd(&maskS[t], sm);
  }
}

// z_hat = x_c@Wz_masked^T + bz ; c_hat = beta*z+(1-beta)*x_hat ; c_c -> outputs; loss2+3
__global__ void k_zc(const float* __restrict__ inp, int t,
                     const float* __restrict__ x_c,
                     const float* __restrict__ Wz, const float* __restrict__ bz,
                     const float* __restrict__ beta, const float* __restrict__ x_hat,
                     float* __restrict__ out, float* __restrict__ lossS) {
  GEMM_SHARED
  int bm = blockIdx.y * BM;
  int nb = blockIdx.x * BN;
  v8f acc[2][2] = {};
  SegDesc s[1] = {{x_c, F_DIM, Wz, F_DIM, F_DIM}};
  gemm_run<true>(s, 1, bm, nb, acc, As0, Ws0, As1, Ws1);  // ZDIAG
  float s23 = 0.f;
  tile_foreach(bm, nb, acc, [&](int row, int col, float a) {
    float zh = a + bz[col];
    float be = beta[row * F_DIM + col];
    float xh = x_hat[row * F_DIM + col];
    float ch = be * zh + (1.f - be) * xh;
    float m  = inp[row * TF3 + 2 * TF + t * F_DIM + col];
    float vv = inp[row * TF3 + t * F_DIM + col];
    float cc = m * vv + (1.f - m) * ch;
    int oo = row * TF + t * F_DIM + col;
    out[oo]       = ch;   // c_hat_list
    out[BTF + oo] = cc;   // imputations
    s23 += (fabsf(vv - zh) + fabsf(vv - ch)) * m;
  });
  atomicAdd(&lossS[t], s23);
}

// cols 0..1535: gl = [c_c, m]@Wih^T + bih ; cols 1536..3071: hl = hid_dec@Whh^T + bhh
__global__ void k_gru(const float* __restrict__ inp, int t,
                      const float* __restrict__ out,      // c_c lives in out[BTF..]
                      const float* __restrict__ hid_dec,
                      const float* __restrict__ Wih, const float* __restrict__ bih,
                      const float* __restrict__ Whh, const float* __restrict__ bhh,
                      float* __restrict__ glbuf, float* __restrict__ hlbuf) {
  GEMM_SHARED
  int bm = blockIdx.y * BM;
  int nb = blockIdx.x * BN;
  v8f acc[2][2] = {};
  if (nb < H3) {
    const float* ccbase = out + BTF + t * F_DIM;  // row stride TF
    const float* mbase  = inp + 2 * TF + t * F_DIM;
    SegDesc s[2] = {{ccbase, TF, Wih, 2 * F_DIM, F_DIM},
                    {mbase, TF3, Wih + F_DIM, 2 * F_DIM, F_DIM}};
    gemm_run<false>(s, 2, bm, nb, acc, As0, Ws0, As1, Ws1);
    tile_foreach(bm, nb, acc, [&](int row, int col, float a) {
      glbuf[row * H3 + col] = a + bih[col];
    });
  } else {
    int nl = nb - H3;
    SegDesc s[1] = {{hid_dec, H_DIM, Whh, H_DIM, H_DIM}};
    gemm_run<false>(s, 1, bm, nl, acc, As0, Ws0, As1, Ws1);
    tile_foreach(bm, nl, acc, [&](int row, int col, float a) {
      hlbuf[row * H3 + col] = a + bhh[col];
    });
  }
}

// GRU elementwise update: hid = (1-z)*n + z*hid_dec
__global__ void k_gate(const float* __restrict__ glbuf, const float* __restrict__ hlbuf,
                       const float* __restrict__ hid_dec, float* __restrict__ hid) {
  int idx = blockIdx.x * blockDim.x + threadIdx.x;
  if (idx >= BB * H_DIM) return;
  int b = idx >> 9;          // /H_DIM
  int j = idx & (H_DIM - 1);
  int base = b * H3;
  float r  = sigmoidf_(glbuf[base + j] + hlbuf[base + j]);
  float z  = sigmoidf_(glbuf[base + H_DIM + j] + hlbuf[base + H_DIM + j]);
  float nn = tanhf(glbuf[base + 2 * H_DIM + j] + r * hlbuf[base + 2 * H_DIM + j]);
  hid[idx] = (1.f - z) * nn + z * hid_dec[idx];
}

__global__ void k_init(float* __restrict__ hid, float* __restrict__ lossS,
                       float* __restrict__ maskS) {
  int i = blockIdx.x * blockDim.x + threadIdx.x;
  if (i < BB * H_DIM) hid[i] = 0.f;
  if (i < TT) { lossS[i] = 0.f; maskS[i] = 0.f; }
}

__global__ void k_final(const float* __restrict__ lossS, const float* __restrict__ maskS,
                        float* __restrict__ out) {
  if (threadIdx.x == 0 && blockIdx.x == 0) {
    float acc = 0.f;
    for (int t = 0; t < TT; ++t) acc += lossS[t] / (maskS[t] + 1e-5f);
    out[2 * BTF] = acc;
  }
}

// ---------------- launch ----------------
extern "C" void kernel_launch(void* const* d_in, const int* in_sizes, int n_in,
                              void* d_out, int out_size, void* d_ws, size_t ws_size,
                              hipStream_t stream) {
  (void)in_sizes; (void)n_in; (void)out_size; (void)ws_size;
  const float* inp   = (const float*)d_in[0];
  const float* Wdx   = (const float*)d_in[1];
  const float* bdx   = (const float*)d_in[2];
  const float* Wdh   = (const float*)d_in[3];
  const float* bdh   = (const float*)d_in[4];
  const float* Wout  = (const float*)d_in[5];
  const float* bout  = (const float*)d_in[6];
  const float* Wz    = (const float*)d_in[7];
  const float* bz    = (const float*)d_in[8];
  const float* Wbeta = (const float*)d_in[9];
  const float* bbeta = (const float*)d_in[10];
  const float* Wih   = (const float*)d_in[11];
  const float* Whh   = (const float*)d_in[12];
  const float* bih   = (const float*)d_in[13];
  const float* bhh   = (const float*)d_in[14];
  float* out = (float*)d_out;

  float* ws      = (float*)d_ws;
  float* hid     = ws;                    // B*H
  float* hid_dec = hid + BB * H_DIM;      // B*H
  float* gamma_x = hid_dec + BB * H_DIM;  // B*F
  float* beta    = gamma_x + BB * F_DIM;  // B*F
  float* x_hat   = beta + BB * F_DIM;     // B*F
  float* x_c     = x_hat + BB * F_DIM;    // B*F
  float* glbuf   = x_c + BB * F_DIM;      // B*3H
  float* hlbuf   = glbuf + BB * H3;       // B*3H
  float* lossS   = hlbuf + BB * H3;       // T
  float* maskS   = lossS + TT;            // T

  k_init<<<(BB * H_DIM + 255) / 256, 256, 0, stream>>>(hid, lossS, maskS);

  for (int t = 0; t < TT; ++t) {
    k_gamma<<<dim3((F_DIM + H_DIM) / BN, BB / BM), NTHREADS, 0, stream>>>(
        inp, t, Wdx, bdx, Wdh, bdh, hid, gamma_x, hid_dec);
    k_beta_xhat<<<dim3((2 * F_DIM) / BN, BB / BM), NTHREADS, 0, stream>>>(
        inp, t, gamma_x, hid_dec, Wbeta, bbeta, Wout, bout, beta, x_hat, x_c, lossS, maskS);
    k_zc<<<dim3(F_DIM / BN, BB / BM), NTHREADS, 0, stream>>>(
        inp, t, x_c, Wz, bz, beta, x_hat, out, lossS);
    k_gru<<<dim3((2 * H3) / BN, BB / BM), NTHREADS, 0, stream>>>(
        inp, t, out, hid_dec, Wih, bih, Whh, bhh, glbuf, hlbuf);
    k_gate<<<(BB * H_DIM + 255) / 256, 256, 0, stream>>>(glbuf, hlbuf, hid_dec, hid);
  }

  k_final<<<1, 32, 0, stream>>>(lossS, maskS, out);
}